// GaussianDDKernel_10299331576074
// MI455X (gfx1250) — compile-verified
//
#include <hip/hip_runtime.h>

typedef __attribute__((ext_vector_type(2))) float v2f;
typedef __attribute__((ext_vector_type(4))) float v4f;
typedef __attribute__((ext_vector_type(8))) float v8f;

#define TILE 128
#define D_DIM 64
#define LDS_STRIDE 68  // 64 + 4 pad: 272B row stride = 17*16B (keeps b128 alignment),
                       // bank = (4*row + k) % 64 -> conflict-free 16x4 fragment loads

__global__ __launch_bounds__(256) void gauss_dd_wmma_kernel(
    const float* __restrict__ X,      // [N, 64]
    const float* __restrict__ Y,      // [M, 64]
    const float* __restrict__ sigma_p,
    float* __restrict__ out,          // [N, M]
    int Mtotal)
{
    __shared__ float Xs[TILE * LDS_STRIDE];
    __shared__ float Ys[TILE * LDS_STRIDE];
    __shared__ float xs2[TILE];
    __shared__ float ys2[TILE];

    const int tid  = threadIdx.x;
    const int lane = tid & 31;
    const int wave = tid >> 5;

    const int rowBase = blockIdx.y * TILE;  // x rows (output rows)
    const int colBase = blockIdx.x * TILE;  // y rows (output cols)

    // ---- Stage X,Y tiles to LDS: 128 rows x 16 float4 each, 8 float4 per thread ----
    {
        const v4f* Xg = (const v4f*)(X + (size_t)rowBase * D_DIM);
        const v4f* Yg = (const v4f*)(Y + (size_t)colBase * D_DIM);
        #pragma unroll
        for (int i = 0; i < 8; ++i) {
            int idx = tid + i * 256;
            int r   = idx >> 4;   // 0..127
            int c4  = idx & 15;   // 0..15
            v4f xv = Xg[r * (D_DIM / 4) + c4];
            v4f yv = Yg[r * (D_DIM / 4) + c4];
            *(v4f*)(&Xs[r * LDS_STRIDE + c4 * 4]) = xv;
            *(v4f*)(&Ys[r * LDS_STRIDE + c4 * 4]) = yv;
        }
    }
    __syncthreads();

    // ---- Per-row sum of squares: thread t -> row t (X for t<128, Y for t>=128) ----
    {
        float s = 0.f;
        const float* src = (tid < 128) ? &Xs[tid * LDS_STRIDE]
                                       : &Ys[(tid - 128) * LDS_STRIDE];
        #pragma unroll
        for (int d = 0; d < D_DIM; ++d) s += src[d] * src[d];
        if (tid < 128) xs2[tid] = s; else ys2[tid - 128] = s;
    }
    __syncthreads();

    // ---- WMMA GEMM: wave -> 32x64 sub-tile = 2x4 tiles of 16x16, K in steps of 4 ----
    const int wr = (wave & 3) * 32;   // wave row offset inside 128x128 tile
    const int wc = (wave >> 2) * 64;  // wave col offset
    const int fr = lane & 15;         // fragment row (A) / col (B)
    const int kh = lane >> 4;         // K half-select: lanes 16-31 hold K+2,K+3

    const v8f vzero = {0.f, 0.f, 0.f, 0.f, 0.f, 0.f, 0.f, 0.f};
    v8f acc[2][4];
    #pragma unroll
    for (int i = 0; i < 2; ++i)
        #pragma unroll
        for (int j = 0; j < 4; ++j) acc[i][j] = vzero;

    for (int k0 = 0; k0 < D_DIM; k0 += 4) {
        v2f a[2], b[4];
        #pragma unroll
        for (int i = 0; i < 2; ++i)
            a[i] = *(const v2f*)(&Xs[(wr + i * 16 + fr) * LDS_STRIDE + k0 + kh * 2]);
        #pragma unroll
        for (int j = 0; j < 4; ++j)
            b[j] = *(const v2f*)(&Ys[(wc + j * 16 + fr) * LDS_STRIDE + k0 + kh * 2]);

        #pragma unroll
        for (int i = 0; i < 2; ++i)
            #pragma unroll
            for (int j = 0; j < 4; ++j)
                acc[i][j] = __builtin_amdgcn_wmma_f32_16x16x4_f32(
                    /*neg_a=*/false, a[i], /*neg_b=*/false, b[j],
                    /*c_mod=*/(short)0, acc[i][j],
                    /*reuse_a=*/false, /*reuse_b=*/false);
    }

    // ---- Epilogue: dist^2 = |x|^2 + |y|^2 - 2*cross; apply d2 RBF; NT stores ----
    const float sig    = *sigma_p;         // uniform -> scalar load
    const float sig2   = sig * sig;
    const float inv_s2 = 1.0f / sig2;
    const float inv_s4 = inv_s2 * inv_s2;
    const float nhalf  = -0.5f * inv_s2;

    #pragma unroll
    for (int i = 0; i < 2; ++i) {
        #pragma unroll
        for (int j = 0; j < 4; ++j) {
            const int n  = wc + j * 16 + fr;       // col inside tile
            const float ysq = ys2[n];
            #pragma unroll
            for (int r = 0; r < 8; ++r) {
                const int m = wr + i * 16 + kh * 8 + r;  // row inside tile (C layout)
                const float cross = acc[i][j][r];
                const float dist  = xs2[m] + ysq - 2.0f * cross;
                const float kern  = __expf(dist * nhalf);
                const float val   = (dist * inv_s4 - inv_s2) * kern;
                const size_t g = (size_t)(rowBase + m) * (size_t)Mtotal + (colBase + n);
                __builtin_nontemporal_store(val, &out[g]);
            }
        }
    }
}

extern "C" void kernel_launch(void* const* d_in, const int* in_sizes, int n_in,
                              void* d_out, int out_size, void* d_ws, size_t ws_size,
                              hipStream_t stream) {
    const float* x     = (const float*)d_in[0];
    const float* y     = (const float*)d_in[1];
    const float* sigma = (const float*)d_in[2];
    float* out         = (float*)d_out;

    const int N = in_sizes[0] / D_DIM;  // 8192
    const int M = in_sizes[1] / D_DIM;  // 8192

    dim3 grid(M / TILE, N / TILE);      // (cols, rows)
    gauss_dd_wmma_kernel<<<grid, 256, 0, stream>>>(x, y, sigma, out, M);
}